// SLDS_73529840107954
// MI455X (gfx1250) — compile-verified
//
#include <hip/hip_runtime.h>
#include <math.h>

#define KC 8
#define DH 64
#define OB 32
#define TT 512

typedef float v2f __attribute__((ext_vector_type(2)));
typedef float v8f __attribute__((ext_vector_type(8)));

static __device__ __forceinline__ v8f vzero8() {
  v8f z = {0.f, 0.f, 0.f, 0.f, 0.f, 0.f, 0.f, 0.f};
  return z;
}

// D[16MB x 16NB] (+)= A[16MB x KT] * B[KT x 16NB], f32 WMMA 16x16x4, row-major A/B.
// A layout (ISA 7.12.2, 32-bit A 16x4): lanes0-15 rows, VGPR0/1 = K0/K1; hi half K2/K3.
// B layout (4x16): VGPR0 = K0 (lo half) / K2 (hi), VGPR1 = K1 / K3.
template <int MB, int NB, int KT>
static __device__ __forceinline__ void wmma_mm(const float* A, int lda,
                                               const float* B, int ldb,
                                               v8f* acc, int lane) {
  const int r16 = lane & 15;
  const int hi  = lane >> 4;
#pragma unroll
  for (int kc = 0; kc < KT; kc += 4) {
    v2f a[MB];
#pragma unroll
    for (int mb = 0; mb < MB; ++mb) {
      const float* ap = A + (mb * 16 + r16) * lda + kc + hi * 2;
      a[mb].x = ap[0];
      a[mb].y = ap[1];
    }
    v2f b[NB];
#pragma unroll
    for (int nb = 0; nb < NB; ++nb) {
      const float* bp = B + (kc + hi * 2) * ldb + nb * 16 + r16;
      b[nb].x = bp[0];
      b[nb].y = bp[ldb];
    }
#pragma unroll
    for (int mb = 0; mb < MB; ++mb)
#pragma unroll
      for (int nb = 0; nb < NB; ++nb)
        acc[mb * NB + nb] = __builtin_amdgcn_wmma_f32_16x16x4_f32(
            false, a[mb], false, b[nb], (short)0, acc[mb * NB + nb], false, false);
  }
}

// C/D 16x16 f32 tile store: VGPR v -> row (hi*8+v), col = lane&15 per tile.
template <int MB, int NB>
static __device__ __forceinline__ void wmma_store(const v8f* acc, float* Out,
                                                  int ldo, int lane) {
  const int r16 = lane & 15;
  const int hi  = lane >> 4;
#pragma unroll
  for (int mb = 0; mb < MB; ++mb)
#pragma unroll
    for (int nb = 0; nb < NB; ++nb)
#pragma unroll
      for (int v = 0; v < 8; ++v)
        Out[(mb * 16 + hi * 8 + v) * ldo + nb * 16 + r16] = acc[mb * NB + nb][v];
}

// Kalman measurement update for wave j. G holds P (64x64) in LDS, Aux[576..639]
// holds the predicted mean, lwnew is the pre-update log-weight. Writes sm[j],
// sLw[j], and G <- posterior covariance. Force-inlined so LDS pointers keep
// addrspace(3) and lower to ds_load/ds_store instead of flat_*.
static __device__ __forceinline__ void obs_update(
    int j, int lane, float* G, float* PCw, float* Ktw, float* Sw, float* Aux,
    float* sm, float* sLw, const float* sR, const float* Cg, const float* CTg,
    const float* y, float lwnew) {
  const int r16 = lane & 15;
  const int hi  = lane >> 4;

  // PC = P @ C  (64x64 @ 64x32)
  v8f pacc[8];
#pragma unroll
  for (int q = 0; q < 8; ++q) pacc[q] = vzero8();
  wmma_mm<4, 2, 64>(G, 64, Cg, 32, pacc, lane);
  wmma_store<4, 2>(pacc, PCw, 32, lane);
  __builtin_amdgcn_wave_barrier();

  // S = C^T @ PC + diag(R)  (32x32), stored ld=33
  v8f sacc[4];
#pragma unroll
  for (int q = 0; q < 4; ++q) sacc[q] = vzero8();
  wmma_mm<2, 2, 64>(CTg, 64, PCw, 32, sacc, lane);
#pragma unroll
  for (int mb = 0; mb < 2; ++mb)
#pragma unroll
    for (int nb = 0; nb < 2; ++nb)
#pragma unroll
      for (int v = 0; v < 8; ++v) {
        int row = mb * 16 + hi * 8 + v;
        int col = nb * 16 + r16;
        float val = sacc[mb * 2 + nb][v];
        val += (row == col) ? sR[row] : 0.f;
        Sw[row * 33 + col] = val;
      }
  __builtin_amdgcn_wave_barrier();

  // In-place Cholesky of S (lower). Lane = row.
  for (int c = 0; c < 32; ++c) {
    float dg = sqrtf(Sw[c * 33 + c]);
    __builtin_amdgcn_wave_barrier();
    if (lane == c) Sw[c * 33 + c] = dg;
    if (lane > c) Sw[lane * 33 + c] = Sw[lane * 33 + c] / dg;
    __builtin_amdgcn_wave_barrier();
    if (lane > c) {
      float lrc = Sw[lane * 33 + c];
      for (int c2 = c + 1; c2 <= lane; ++c2)
        Sw[lane * 33 + c2] -= lrc * Sw[c2 * 33 + c];
    }
    __builtin_amdgcn_wave_barrier();
  }

  // innovation v = y - m_pred @ C (lane = obs index)
  {
    float mu = 0.f;
    for (int a = 0; a < DH; ++a) mu += Aux[576 + a] * Cg[a * 32 + lane];
    Aux[640 + lane] = y[lane] - mu;
  }
  __builtin_amdgcn_wave_barrier();

  // w = L^{-1} v, log-likelihood (redundant across lanes)
  float w[32];
  float sumw2 = 0.f, sumlog = 0.f;
#pragma unroll
  for (int k = 0; k < 32; ++k) {
    float s = Aux[640 + k];
#pragma unroll
    for (int p = 0; p < k; ++p) s -= Sw[k * 33 + p] * w[p];
    float dk = Sw[k * 33 + k];
    w[k] = s / dk;
    sumw2 += w[k] * w[k];
    sumlog += logf(dk);
  }
  const float LOG2PI = 1.8378770664093453f;
  float newLw = lwnew + (-0.5f * sumw2 - sumlog - 0.5f * 32.0f * LOG2PI);

  // Kt = S^{-1} PC^T via two triangular solves; lane owns RHS cols lane, lane+32.
  float x0[32], x1[32];
  const int c0 = lane, c1 = lane + 32;
#pragma unroll
  for (int k = 0; k < 32; ++k) {
    float s0 = PCw[c0 * 32 + k];
    float s1 = PCw[c1 * 32 + k];
#pragma unroll
    for (int p = 0; p < k; ++p) {
      float L = Sw[k * 33 + p];
      s0 -= L * x0[p];
      s1 -= L * x1[p];
    }
    float dk = Sw[k * 33 + k];
    x0[k] = s0 / dk;
    x1[k] = s1 / dk;
  }
#pragma unroll
  for (int k = 31; k >= 0; --k) {
    float s0 = x0[k], s1 = x1[k];
#pragma unroll
    for (int p = k + 1; p < 32; ++p) {
      float L = Sw[p * 33 + k];
      s0 -= L * x0[p];
      s1 -= L * x1[p];
    }
    float dk = Sw[k * 33 + k];
    x0[k] = s0 / dk;
    x1[k] = s1 / dk;
  }
#pragma unroll
  for (int k = 0; k < 32; ++k) {
    Ktw[k * 64 + c0] = x0[k];
    Ktw[k * 64 + c1] = x1[k];
  }
  __builtin_amdgcn_wave_barrier();

  // m <- m_pred + v @ Kt  (lane handles dims lane, lane+32)
  {
    float m0 = Aux[576 + lane], m1 = Aux[576 + lane + 32];
    for (int o = 0; o < 32; ++o) {
      float vo = Aux[640 + o];
      m0 += vo * Ktw[o * 64 + lane];
      m1 += vo * Ktw[o * 64 + lane + 32];
    }
    sm[j * 64 + lane] = m0;
    sm[j * 64 + lane + 32] = m1;
  }

  // P <- P - PC @ Kt  (64x32 @ 32x64)
  v8f uacc[16];
#pragma unroll
  for (int q = 0; q < 16; ++q) uacc[q] = vzero8();
  wmma_mm<4, 4, 32>(PCw, 32, Ktw, 64, uacc, lane);
#pragma unroll
  for (int mb = 0; mb < 4; ++mb)
#pragma unroll
    for (int nb = 0; nb < 4; ++nb)
#pragma unroll
      for (int v = 0; v < 8; ++v) {
        int idx = (mb * 16 + hi * 8 + v) * 64 + nb * 16 + r16;
        G[idx] = G[idx] - uacc[mb * 4 + nb][v];
      }

  if (lane == 0) sLw[j] = newLw;
}

// Shared memory layout (floats): sP 32768 | PC 16384 | Kt 16384 | S 8448 |
// Aux 5632 | lw 8 | logA 64 | Q 64 | R 32 | m 512  => 80296 f = 321184 B (<320KB)
#define SH_FLOATS 80296

__global__ __launch_bounds__(256, 1) void slds_main(
    const float* __restrict__ data, const float* __restrict__ tl,
    const float* __restrict__ M, const float* __restrict__ ltn,
    const float* __restrict__ Cg, const float* __restrict__ lon,
    const float* __restrict__ MT, const float* __restrict__ CTg, float* out) {
  extern __shared__ float smem[];
  float* sP    = smem;               // 8 x 64 x 64 state covariances / scratch
  float* sPCa  = sP + 32768;
  float* sKta  = sPCa + 16384;
  float* sSa   = sKta + 16384;
  float* sAxa  = sSa + 8448;
  float* sLw   = sAxa + 5632;
  float* slogA = sLw + 8;
  float* sQ    = slogA + 64;
  float* sR    = sQ + 64;
  float* sm    = sR + 32;

  const int tid  = threadIdx.x;
  const int lane = tid & 31;
  const int j    = tid >> 5;
  const int r16  = lane & 15;
  const int hi   = lane >> 4;

  // Constants: logA = log_softmax rows, Q/R diag exponentials.
  if (tid < 8) {
    float mx = -INFINITY;
    for (int c = 0; c < 8; ++c) mx = fmaxf(mx, tl[tid * 8 + c]);
    float s = 0.f;
    for (int c = 0; c < 8; ++c) s += expf(tl[tid * 8 + c] - mx);
    float lse = mx + logf(s);
    for (int c = 0; c < 8; ++c) slogA[tid * 8 + c] = tl[tid * 8 + c] - lse;
  }
  if (tid < 64) sQ[tid] = expf(ltn[tid]);
  if (tid < 32) sR[tid] = expf(lon[tid]);

  float* G   = sP + j * 4096;
  float* PCw = sPCa + j * 2048;
  float* Ktw = sKta + j * 2048;
  float* Sw  = sSa + j * 1056;
  float* Aux = sAxa + j * 704;  // [0..511] d_i, [512..575] mbar, [576..639] m_pred, [640..671] v

  // t=0 init: P = I, m = 0, m_pred = 0
  for (int v = 0; v < 128; ++v) {
    int idx = v * 32 + lane;
    int row = idx >> 6, col = idx & 63;
    G[idx] = (row == col) ? 1.f : 0.f;
  }
  sm[j * 64 + lane] = 0.f;
  sm[j * 64 + lane + 32] = 0.f;
  Aux[576 + lane] = 0.f;
  Aux[576 + lane + 32] = 0.f;
  __syncthreads();

  for (int t = 0; t < TT; ++t) {
    // Prefetch next observation row into cache (CDNA5 global_prefetch path).
    if (t + 1 < TT) __builtin_prefetch(data + (t + 1) * OB, 0, 1);

    float lwnew;
    if (t == 0) {
      lwnew = slogA[j];  // s_0 conditioned on previous component 0
    } else {
      __syncthreads();
      // ---- Phase A: read shared state (lw, m, P) ----
      float r[8];
      {
        float lm[8], mx = -INFINITY;
#pragma unroll
        for (int i = 0; i < 8; ++i) {
          lm[i] = sLw[i] + slogA[i * 8 + j];
          mx = fmaxf(mx, lm[i]);
        }
        float s = 0.f;
#pragma unroll
        for (int i = 0; i < 8; ++i) s += expf(lm[i] - mx);
        lwnew = mx + logf(s);
#pragma unroll
        for (int i = 0; i < 8; ++i) r[i] = expf(lm[i] - lwnew);
      }
      // mbar = sum_i r_i m_i ; d_i = m_i - mbar
      {
        float mb0 = 0.f, mb1 = 0.f;
#pragma unroll
        for (int i = 0; i < 8; ++i) {
          mb0 += r[i] * sm[i * 64 + lane];
          mb1 += r[i] * sm[i * 64 + lane + 32];
        }
        Aux[512 + lane] = mb0;
        Aux[512 + lane + 32] = mb1;
#pragma unroll
        for (int i = 0; i < 8; ++i) {
          Aux[i * 64 + lane] = sm[i * 64 + lane] - mb0;
          Aux[i * 64 + lane + 32] = sm[i * 64 + lane + 32] - mb1;
        }
      }
      __builtin_amdgcn_wave_barrier();
      // m_pred = mbar @ M[j]
      {
        const float* Mj = M + j * 4096;
        float mn0 = 0.f, mn1 = 0.f;
        for (int a = 0; a < 64; ++a) {
          float mba = Aux[512 + a];
          mn0 += mba * Mj[a * 64 + lane];
          mn1 += mba * Mj[a * 64 + lane + 32];
        }
        Aux[576 + lane] = mn0;
        Aux[576 + lane + 32] = mn1;
      }
      // Pbar = sum_i r_i (P_i + d_i d_i^T) accumulated straight into WMMA C/D layout.
      v8f acc[16];
#pragma unroll
      for (int q = 0; q < 16; ++q) acc[q] = vzero8();
#pragma unroll
      for (int mb = 0; mb < 4; ++mb)
#pragma unroll
        for (int v = 0; v < 8; ++v) {
          int row = mb * 16 + hi * 8 + v;
#pragma unroll
          for (int nb = 0; nb < 4; ++nb) {
            int col = nb * 16 + r16;
            float s = 0.f;
#pragma unroll
            for (int i = 0; i < 8; ++i)
              s += r[i] * (sP[i * 4096 + row * 64 + col] +
                           Aux[i * 64 + row] * Aux[i * 64 + col]);
            acc[mb * 4 + nb][v] = s;
          }
        }
      __syncthreads();
      // ---- Phase B: own-region compute ----
      wmma_store<4, 4>(acc, G, 64, lane);  // G <- Pbar
      __builtin_amdgcn_wave_barrier();

      // W = Pbar @ M[j]
#pragma unroll
      for (int q = 0; q < 16; ++q) acc[q] = vzero8();
      wmma_mm<4, 4, 64>(G, 64, M + j * 4096, 64, acc, lane);
      wmma_store<4, 4>(acc, G, 64, lane);  // G <- W
      __builtin_amdgcn_wave_barrier();

      // P_pred = M[j]^T @ W + diag(Q)
#pragma unroll
      for (int q = 0; q < 16; ++q) acc[q] = vzero8();
      wmma_mm<4, 4, 64>(MT + j * 4096, 64, G, 64, acc, lane);
#pragma unroll
      for (int mb = 0; mb < 4; ++mb)
#pragma unroll
        for (int v = 0; v < 8; ++v) {
          int row = mb * 16 + hi * 8 + v;
          acc[mb * 4 + mb][v] += (r16 == hi * 8 + v) ? sQ[row] : 0.f;
        }
      wmma_store<4, 4>(acc, G, 64, lane);  // G <- P_pred
      __builtin_amdgcn_wave_barrier();
    }

    obs_update(j, lane, G, PCw, Ktw, Sw, Aux, sm, sLw, sR, Cg, CTg,
               data + t * OB, lwnew);
  }
  __syncthreads();
  if (tid == 0) {
    float mx = -INFINITY;
    for (int i = 0; i < 8; ++i) mx = fmaxf(mx, sLw[i]);
    float s = 0.f;
    for (int i = 0; i < 8; ++i) s += expf(sLw[i] - mx);
    out[0] = mx + logf(s);
  }
}

// Materialize M^T (per component) and C^T in workspace so WMMA A-operands are
// row-major contiguous.
__global__ void slds_prep(const float* __restrict__ M, const float* __restrict__ Cg,
                          float* __restrict__ MT, float* __restrict__ CTg) {
  int idx = blockIdx.x * blockDim.x + threadIdx.x;
  if (idx < KC * DH * DH) {
    int k = idx >> 12;
    int a = (idx >> 6) & 63;
    int d = idx & 63;
    MT[k * 4096 + d * 64 + a] = M[idx];
  }
  if (idx < OB * DH) {
    int o = idx >> 6;
    int a = idx & 63;
    CTg[o * 64 + a] = Cg[a * 32 + o];
  }
}

extern "C" void kernel_launch(void* const* d_in, const int* in_sizes, int n_in,
                              void* d_out, int out_size, void* d_ws, size_t ws_size,
                              hipStream_t stream) {
  (void)in_sizes; (void)n_in; (void)out_size; (void)ws_size;
  const float* data = (const float*)d_in[0];  // (512, 32)
  const float* tl   = (const float*)d_in[1];  // (8, 8)
  const float* M    = (const float*)d_in[2];  // (8, 64, 64)
  const float* ltn  = (const float*)d_in[3];  // (64,)
  const float* Cg   = (const float*)d_in[4];  // (64, 32)
  const float* lon  = (const float*)d_in[5];  // (32,)
  float* MT = (float*)d_ws;                   // 8*64*64 floats
  float* CT = MT + KC * DH * DH;              // 32*64 floats
  float* out = (float*)d_out;

  slds_prep<<<(KC * DH * DH + 255) / 256, 256, 0, stream>>>(M, Cg, MT, CT);

  const size_t shbytes = (size_t)SH_FLOATS * sizeof(float);
  hipFuncSetAttribute((const void*)slds_main,
                      hipFuncAttributeMaxDynamicSharedMemorySize, (int)shbytes);
  slds_main<<<1, 256, shbytes, stream>>>(data, tl, M, ltn, Cg, lon, MT, CT, out);
}